// DensityAndFeaturesMLP_48232482734138
// MI455X (gfx1250) — compile-verified
//
#include <hip/hip_runtime.h>
#include <hip/hip_bf16.h>

typedef __attribute__((ext_vector_type(2))) float v2f;
typedef __attribute__((ext_vector_type(8))) float v8f;

#define NLEV 20
#define NDENSE 7            // levels 0..6 are dense, 7..19 hashed (params = 2^21)
#define HASH_MASK 0x1FFFFFu

// res_l = ceil(16 * SCALE^l), SCALE = exp((ln 8193 - ln 16)/19)
__device__ __constant__ unsigned c_res[NLEV] = {
    16u, 23u, 31u, 43u, 60u, 83u, 115u, 160u, 222u, 308u,
    427u, 593u, 823u, 1143u, 1587u, 2204u, 3060u, 4249u, 5900u, 8193u};
// running offsets of 8-aligned per-level param counts
__device__ __constant__ unsigned c_off[NLEV] = {
    0u, 4920u, 18744u, 51512u, 136696u, 363680u, 956384u, 2517280u,
    4614432u, 6711584u, 8808736u, 10905888u, 13003040u, 15100192u,
    17197344u, 19294496u, 21391648u, 23488800u, 25585952u, 27683104u};

#define BLOCK 128
#define FSTRIDE 41          // 40 feats + 1 pad (odd -> bank-conflict-free)
#define HSTRIDE 66          // 64 hidden + 2 pad (write halves hit disjoint banks)

__global__ __launch_bounds__(BLOCK)
void ngp_fused_kernel(const float* __restrict__ pos,
                      const int* __restrict__ boundp,
                      const float* __restrict__ table,
                      const float* __restrict__ w1,
                      const float* __restrict__ wd,
                      const float* __restrict__ wf,
                      float* __restrict__ out,
                      int npts)
{
    __shared__ float s_feat[BLOCK * FSTRIDE];          // [128][40] features
    __shared__ float s_w1[64 * FSTRIDE];               // w1[n][k] at n*41+k
    __shared__ float s_b2[64 * 16];                    // B2[k][n]: n<7 -> wf, n==7 -> wd, else 0
    __shared__ float s_h[(BLOCK / 32) * 16 * HSTRIDE]; // per-wave hidden tile

    const int tid  = threadIdx.x;
    const int lane = tid & 31;
    const int wave = tid >> 5;
    const int blk0 = blockIdx.x * BLOCK;

    // ---- preload weights into LDS ----
    for (int i = tid; i < 64 * 40; i += BLOCK) {
        int n = i / 40, k = i - n * 40;
        s_w1[n * FSTRIDE + k] = w1[i];
    }
    for (int i = tid; i < 64 * 16; i += BLOCK) {
        int k = i >> 4, n = i & 15;
        float v = 0.0f;
        if (n < 7)       v = wf[n * 64 + k];
        else if (n == 7) v = wd[k];
        s_b2[i] = v;
    }

    // ---- phase 1: hashgrid encode, one point per thread ----
    const float bnd   = (float)boundp[0];
    const float inv2b = 1.0f / (2.0f * bnd);
    const int   pi    = blk0 + tid;
    float px = 0.f, py = 0.f, pz = 0.f;
    if (pi < npts) {
        px = (pos[pi * 3 + 0] + bnd) * inv2b;
        py = (pos[pi * 3 + 1] + bnd) * inv2b;
        pz = (pos[pi * 3 + 2] + bnd) * inv2b;
    }
    for (int l = 0; l < NLEV; ++l) {
        const unsigned res = c_res[l];
        const float fr = (float)res;
        const float fx = px * fr, fy = py * fr, fz = pz * fr;
        float x0 = fminf(fmaxf(floorf(fx), 0.f), fr - 1.f);
        float y0 = fminf(fmaxf(floorf(fy), 0.f), fr - 1.f);
        float z0 = fminf(fmaxf(floorf(fz), 0.f), fr - 1.f);
        const float frx = fx - x0, fry = fy - y0, frz = fz - z0;
        const float omx = 1.f - frx, omy = 1.f - fry, omz = 1.f - frz;
        const unsigned ux = (unsigned)x0, uy = (unsigned)y0, uz = (unsigned)z0;
        const unsigned s = res + 1u;
        const unsigned base = c_off[l];
        const bool dense = (l < NDENSE);
        float f0 = 0.f, f1 = 0.f;
#pragma unroll
        for (int c = 0; c < 8; ++c) {
            unsigned cx = ux + (unsigned)(c & 1);
            unsigned cy = uy + (unsigned)((c >> 1) & 1);
            unsigned cz = uz + (unsigned)((c >> 2) & 1);
            cx = cx > res ? res : cx;
            cy = cy > res ? res : cy;
            cz = cz > res ? res : cz;
            unsigned idx;
            if (dense) {
                idx = cx + cy * s + cz * s * s;
            } else {
                idx = cx ^ (cy * 2654435761u) ^ (cz * 805459861u);
                idx &= HASH_MASK;
            }
            const float w = ((c & 1) ? frx : omx) *
                            (((c >> 1) & 1) ? fry : omy) *
                            (((c >> 2) & 1) ? frz : omz);
            const v2f e = *(const v2f*)(table + 2u * (base + idx));
            f0 += w * e[0];
            f1 += w * e[1];
        }
        s_feat[tid * FSTRIDE + 2 * l]     = f0;
        s_feat[tid * FSTRIDE + 2 * l + 1] = f1;
    }
    __syncthreads();

    // ---- phase 2: fused MLP via f32 WMMA; wave owns 2 row-tiles of 16 points ----
    const int hi = lane >> 4;   // half-wave: selects K pair (A/B) and M half (C/D)
    const int lm = lane & 15;
    float* hbuf = &s_h[wave * 16 * HSTRIDE];

    for (int mt = 0; mt < 2; ++mt) {
        const int mbase = wave * 32 + mt * 16;
        v8f acc0 = {}, acc1 = {}, acc2 = {}, acc3 = {};
#pragma unroll
        for (int kt = 0; kt < 10; ++kt) {
            const int k = kt * 4 + hi * 2;
            v2f av;
            av[0] = s_feat[(mbase + lm) * FSTRIDE + k];
            av[1] = s_feat[(mbase + lm) * FSTRIDE + k + 1];
            v2f b0, b1, b2, b3;
            b0[0] = s_w1[(lm)      * FSTRIDE + k]; b0[1] = s_w1[(lm)      * FSTRIDE + k + 1];
            b1[0] = s_w1[(lm + 16) * FSTRIDE + k]; b1[1] = s_w1[(lm + 16) * FSTRIDE + k + 1];
            b2[0] = s_w1[(lm + 32) * FSTRIDE + k]; b2[1] = s_w1[(lm + 32) * FSTRIDE + k + 1];
            b3[0] = s_w1[(lm + 48) * FSTRIDE + k]; b3[1] = s_w1[(lm + 48) * FSTRIDE + k + 1];
            acc0 = __builtin_amdgcn_wmma_f32_16x16x4_f32(false, av, false, b0, (short)0, acc0, false, false);
            acc1 = __builtin_amdgcn_wmma_f32_16x16x4_f32(false, av, false, b1, (short)0, acc1, false, false);
            acc2 = __builtin_amdgcn_wmma_f32_16x16x4_f32(false, av, false, b2, (short)0, acc2, false, false);
            acc3 = __builtin_amdgcn_wmma_f32_16x16x4_f32(false, av, false, b3, (short)0, acc3, false, false);
        }
        // ReLU + park hidden tile in LDS (D-layout -> A-layout reshuffle)
#pragma unroll
        for (int r = 0; r < 8; ++r) {
            const int m = r + 8 * hi;
            hbuf[m * HSTRIDE + lm +  0] = fmaxf(acc0[r], 0.f);
            hbuf[m * HSTRIDE + lm + 16] = fmaxf(acc1[r], 0.f);
            hbuf[m * HSTRIDE + lm + 32] = fmaxf(acc2[r], 0.f);
            hbuf[m * HSTRIDE + lm + 48] = fmaxf(acc3[r], 0.f);
        }
        // GEMM2: h[16x64] x B2[64x16] (cols 0..6 = features, 7 = density)
        v8f o = {};
#pragma unroll
        for (int kt = 0; kt < 16; ++kt) {
            const int k = kt * 4 + hi * 2;
            v2f av, bv;
            av[0] = hbuf[lm * HSTRIDE + k];
            av[1] = hbuf[lm * HSTRIDE + k + 1];
            bv[0] = s_b2[k * 16 + lm];
            bv[1] = s_b2[(k + 1) * 16 + lm];
            o = __builtin_amdgcn_wmma_f32_16x16x4_f32(false, av, false, bv, (short)0, o, false, false);
        }
        // d_out = features [N,7] flat, then density [N,1]
#pragma unroll
        for (int r = 0; r < 8; ++r) {
            const int m = r + 8 * hi;
            const int point = blk0 + mbase + m;
            if (point < npts) {
                if (lm < 7)       out[point * 7 + lm] = o[r];
                else if (lm == 7) out[(size_t)npts * 7 + point] = o[r];
            }
        }
    }
}

extern "C" void kernel_launch(void* const* d_in, const int* in_sizes, int n_in,
                              void* d_out, int out_size, void* d_ws, size_t ws_size,
                              hipStream_t stream) {
    (void)n_in; (void)out_size; (void)d_ws; (void)ws_size;
    const float* pos   = (const float*)d_in[0];
    const int*   bound = (const int*)d_in[1];
    const float* table = (const float*)d_in[2];
    const float* w1    = (const float*)d_in[3];
    const float* wd    = (const float*)d_in[4];
    const float* wf    = (const float*)d_in[5];
    float* out = (float*)d_out;

    const int npts = in_sizes[0] / 3;
    const int nblk = (npts + BLOCK - 1) / BLOCK;
    ngp_fused_kernel<<<nblk, BLOCK, 0, stream>>>(pos, bound, table, w1, wd, wf, out, npts);
}